// BilinearAttentionFusion_36129264894658
// MI455X (gfx1250) — compile-verified
//
#include <hip/hip_runtime.h>
#include <cstddef>

// CDNA5 / gfx1250: wave32, WMMA f32 16x16x4.
typedef float v2f __attribute__((ext_vector_type(2)));
typedef float v8f __attribute__((ext_vector_type(8)));

#define BS 32
#define LA 300
#define LB 1024
#define H  512
#define HM (H * H)   // 262144 rows of the big GEMM

// ---------------------------------------------------------------------------
// K1: b_mean[b][e] = (1/LB) * sum_j feat_b[b][j][e]
// Coalesced along e; 64 MB of reads total.
// ---------------------------------------------------------------------------
__global__ void k_bmean(const float* __restrict__ fb, float* __restrict__ bm) {
    int idx = blockIdx.x * blockDim.x + threadIdx.x;   // [0, BS*H)
    int b = idx >> 9;
    int e = idx & (H - 1);
    const float* p = fb + (size_t)b * LB * H + e;
    float s0 = 0.f, s1 = 0.f, s2 = 0.f, s3 = 0.f;
    for (int j = 0; j < LB; j += 4) {
        s0 += p[(size_t)(j + 0) * H];
        s1 += p[(size_t)(j + 1) * H];
        s2 += p[(size_t)(j + 2) * H];
        s3 += p[(size_t)(j + 3) * H];
    }
    bm[idx] = (s0 + s1 + s2 + s3) * (1.0f / LB);
}

// ---------------------------------------------------------------------------
// K2: t[b][m] = sum_e W[m][e] * b_mean[b][e]   (m = k*512+d, flat)
// GEMM M=262144, N=32, K=512 via v_wmma_f32_16x16x4_f32.
// One wave per 16-row W tile; two accumulators cover all 32 batches.
// HBM-bound on the 537 MB stream of W.
// ---------------------------------------------------------------------------
__global__ void k_proj(const float* __restrict__ W, const float* __restrict__ bm,
                       float* __restrict__ t) {
    // [32 batches][16 m] transpose buffer per wave; 20-float row pitch keeps
    // 16B alignment for float4 reads and spreads LDS banks.
    __shared__ __align__(16) float tbuf[8][32][20];

    const int wave = threadIdx.x >> 5;
    const int lane = threadIdx.x & 31;
    const int lo   = lane & 15;
    const int hi   = lane >> 4;
    const int koff = hi * 2;          // A/B frag K-offset per ISA layout

    const size_t m0 = ((size_t)blockIdx.x * 8 + wave) * 16;

    const float* arow = W  + (m0 + lo) * (size_t)H + koff;  // A: 16 rows of W
    const float* b0   = bm + (size_t)lo * H + koff;         // B cols: batches 0..15
    const float* b1   = bm + (size_t)(16 + lo) * H + koff;  // B cols: batches 16..31

    v8f c0 = {};
    v8f c1 = {};
    for (int e = 0; e < H; e += 4) {
        v2f a  = *(const v2f*)(arow + e);
        v2f p0 = *(const v2f*)(b0 + e);
        v2f p1 = *(const v2f*)(b1 + e);
        c0 = __builtin_amdgcn_wmma_f32_16x16x4_f32(false, a, false, p0, (short)0, c0, false, false);
        c1 = __builtin_amdgcn_wmma_f32_16x16x4_f32(false, a, false, p1, (short)0, c1, false, false);
    }

    // D layout: VGPR v -> M = v + 8*hi, N = lo. Stage into LDS to transpose so
    // each lane stores 16 contiguous t elements for one batch (b128 stores).
    #pragma unroll
    for (int v = 0; v < 8; ++v) {
        tbuf[wave][lo][v + 8 * hi]      = c0[v];
        tbuf[wave][16 + lo][v + 8 * hi] = c1[v];
    }
    __syncthreads();

    float* dst = t + (size_t)lane * HM + m0;   // t[batch=lane][m0..m0+15]
    #pragma unroll
    for (int j = 0; j < 16; j += 4) {
        float4 q = { tbuf[wave][lane][j],     tbuf[wave][lane][j + 1],
                     tbuf[wave][lane][j + 2], tbuf[wave][lane][j + 3] };
        *(float4*)(dst + j) = q;
    }
}

// ---------------------------------------------------------------------------
// K3: fused[b][a][k] = sum_d feat_a[b][a][d] * t[b][k][d] + bias[k]
//     x = fused + feat_a ; LayerNorm over k ; * gamma + beta
// Block = (a-tile of 16 rows, batch b); 8 waves x 4 column-tiles = 512 cols.
// LN fused via LDS x-tile + wave32 shuffle reduction.
// ---------------------------------------------------------------------------
__global__ void k_fuse_ln(const float* __restrict__ fa, const float* __restrict__ t,
                          const float* __restrict__ bias, const float* __restrict__ gamma,
                          const float* __restrict__ beta, float* __restrict__ out) {
    __shared__ float xt[16][H];   // 32 KB x-tile

    const int wave = threadIdx.x >> 5;
    const int lane = threadIdx.x & 31;
    const int lo   = lane & 15;
    const int hi   = lane >> 4;
    const int koff = hi * 2;

    const int a0 = blockIdx.x * 16;
    const int b  = blockIdx.y;
    const int arow = a0 + lo;

    const float* fab = fa + (size_t)b * LA * H;
    const float* tb  = t  + (size_t)b * HM;

    v8f c[4];
    #pragma unroll
    for (int q = 0; q < 4; ++q) {
        int k0 = (wave * 4 + q) * 16;
        float bk = bias[k0 + lo];
        v8f init = {bk, bk, bk, bk, bk, bk, bk, bk};   // fold bias into C
        c[q] = init;
    }

    const bool arow_ok = (arow < LA);
    const float* ap = fab + (size_t)arow * H + koff;

    for (int e = 0; e < H; e += 4) {
        v2f a = {0.f, 0.f};
        if (arow_ok) a = *(const v2f*)(ap + e);        // zero-pad tail rows
        #pragma unroll
        for (int q = 0; q < 4; ++q) {
            int k0 = (wave * 4 + q) * 16;
            v2f bv = *(const v2f*)(tb + (size_t)(k0 + lo) * H + e + koff);
            c[q] = __builtin_amdgcn_wmma_f32_16x16x4_f32(false, a, false, bv, (short)0, c[q], false, false);
        }
    }

    // Epilogue: residual add, stage x into LDS.
    #pragma unroll
    for (int q = 0; q < 4; ++q) {
        int k = (wave * 4 + q) * 16 + lo;
        #pragma unroll
        for (int v = 0; v < 8; ++v) {
            int r  = v + 8 * hi;          // row within tile
            int ga = a0 + r;
            float x = c[q][v];
            if (ga < LA) x += fab[(size_t)ga * H + k];
            xt[r][k] = x;
        }
    }
    __syncthreads();

    // LayerNorm: each wave normalizes rows {wave, wave+8}.
    for (int rr = wave; rr < 16; rr += 8) {
        int ga = a0 + rr;
        float s1 = 0.f, s2 = 0.f;
        float vals[16];
        #pragma unroll
        for (int i = 0; i < 16; ++i) {
            float x = xt[rr][lane + i * 32];
            vals[i] = x;
            s1 += x;
            s2 += x * x;
        }
        #pragma unroll
        for (int off = 16; off >= 1; off >>= 1) {
            s1 += __shfl_xor(s1, off, 32);
            s2 += __shfl_xor(s2, off, 32);
        }
        float mu  = s1 * (1.0f / H);
        float var = s2 * (1.0f / H) - mu * mu;
        float rs  = rsqrtf(var + 1e-5f);
        if (ga < LA) {
            float* op = out + ((size_t)b * LA + ga) * H;
            #pragma unroll
            for (int i = 0; i < 16; ++i) {
                int k = lane + i * 32;
                op[k] = (vals[i] - mu) * rs * gamma[k] + beta[k];
            }
        }
    }
}

// ---------------------------------------------------------------------------
extern "C" void kernel_launch(void* const* d_in, const int* in_sizes, int n_in,
                              void* d_out, int out_size, void* d_ws, size_t ws_size,
                              hipStream_t stream) {
    const float* fa    = (const float*)d_in[0];  // [32,300,512]
    const float* fb    = (const float*)d_in[1];  // [32,1024,512]
    const float* W     = (const float*)d_in[2];  // [512,512,512]
    const float* bias  = (const float*)d_in[3];  // [512]
    const float* gamma = (const float*)d_in[4];  // [512]
    const float* beta  = (const float*)d_in[5];  // [512]
    float* out = (float*)d_out;                  // [32,300,512]

    float* bm = (float*)d_ws;                    // 32*512 floats
    float* t  = bm + (size_t)BS * H;             // 32*512*512 floats (~33.5 MB)

    k_bmean<<<dim3((BS * H) / 256), dim3(256), 0, stream>>>(fb, bm);
    k_proj<<<dim3(HM / (16 * 8)), dim3(256), 0, stream>>>(W, bm, t);
    k_fuse_ln<<<dim3((LA + 15) / 16, BS), dim3(256), 0, stream>>>(fa, t, bias, gamma, beta, out);
}